// aggregator_46540265619762
// MI455X (gfx1250) — compile-verified
//
#include <hip/hip_runtime.h>

// CDNA5 / gfx1250 wave32 kernel for deduplicated neighborhood mean aggregation.
//   out[b, :] = (1/|U_b|) * sum_{i in U_b} features[i, :],  U_b = unique({nbr[b,:], node[b]})
//
// Design: 1 block (256 threads = 8 wave32) per node.
//   - wave 0 computes dedup weights with __shfl/__ballot (DEG=32 == wave32 lanes)
//   - 33 feature rows staged to LDS with coalesced float4 loads (1KB burst per row)
//   - weighted sum computed with v_wmma_f32_16x16x4_f32:
//       A (16x4, MxK) = X^T tile: A[m,k] = tile[k][colbase+m]
//         lane 0-15: M=lane, VGPR0=K0, VGPR1=K1 ; lane 16-31: M=lane-16, VGPR0=K2, VGPR1=K3
//       B (4x16)     = weights broadcast over N (constant per K row)
//       C/D (16x16)  : VGPR r holds M=r (lanes 0-15) / M=8+r (lanes 16-31), N=lane%16
//   - lanes 0/16 of each wave hold the N=0 output column -> stage to LDS -> coalesced store

typedef float v2f __attribute__((ext_vector_type(2)));
typedef float v8f __attribute__((ext_vector_type(8)));

#define DEG   32
#define FDIM  256
#define KPAD  36   // 33 neighbor+self slots padded to multiple of 4

__global__ __launch_bounds__(256)
void gnn_dedup_mean_wmma_kernel(const int* __restrict__ nodes,
                                const int* __restrict__ nbr,
                                const float* __restrict__ features,
                                float* __restrict__ out)
{
    __shared__ float tile[KPAD * FDIM];   // 36 KB staged rows (zero-padded K)
    __shared__ float wLds[KPAD];          // dedup weights (0/1), zero padding
    __shared__ int   idxLds[DEG + 1];     // 33 gather indices
    __shared__ float outLds[FDIM];        // result staging for coalesced store
    __shared__ float invDenLds;

    const int b    = blockIdx.x;
    const int t    = threadIdx.x;
    const int lane = t & 31;
    const int wave = t >> 5;

    // ---- wave 0: dedup weights (one neighbor per lane; wave32 == DEG) ----
    if (wave == 0) {
        const int node  = nodes[b];
        const int myidx = nbr[b * DEG + lane];

        bool dupBefore = false;
        #pragma unroll
        for (int j = 0; j < 32; ++j) {
            const int vj = __shfl(myidx, j, 32);
            if (j < lane && vj == myidx) dupBefore = true;
        }
        const unsigned long long uniqMask = __ballot(!dupBefore);
        const unsigned long long selfMask = __ballot(myidx == node);

        wLds[lane]   = dupBefore ? 0.0f : 1.0f;
        idxLds[lane] = myidx;
        if (lane == 0) {
            const int selfUnique = (selfMask == 0ull) ? 1 : 0;
            wLds[32]   = (float)selfUnique;
            idxLds[32] = node;
            wLds[33] = 0.0f; wLds[34] = 0.0f; wLds[35] = 0.0f;
            const int den = __popcll(uniqMask) + selfUnique;
            invDenLds = 1.0f / (float)den;
        }
    }
    __syncthreads();

    // ---- stage rows to LDS: 4 rows in flight, 64 lanes x float4 per row (1KB burst) ----
    {
        const int sub = t >> 6;        // 0..3 : row within group of 4
        const int c   = (t & 63) * 4;  // column base for this thread's float4
        #pragma unroll
        for (int r4 = 0; r4 < KPAD; r4 += 4) {
            const int r = r4 + sub;
            float4 v = make_float4(0.0f, 0.0f, 0.0f, 0.0f);
            if (r < DEG + 1) {
                const long long gi = (long long)idxLds[r] * FDIM + c;
                v = *(const float4*)(features + gi);
            }
            *(float4*)(&tile[r * FDIM + c]) = v;
        }
    }
    __syncthreads();

    // ---- WMMA einsum: out[col] = sum_k w[k] * tile[k][col], f32 16x16x4 ----
    const int m     = lane & 15;             // M index (feature column within block)
    const int khalf = (lane < 16) ? 0 : 2;   // lane-half K split per ISA A/B layout
    #pragma unroll
    for (int cbi = 0; cbi < 2; ++cbi) {
        const int colbase = (wave * 2 + cbi) * 16;
        v8f acc = {0.f, 0.f, 0.f, 0.f, 0.f, 0.f, 0.f, 0.f};
        #pragma unroll
        for (int ks = 0; ks < KPAD / 4; ++ks) {
            const int kslot = ks * 4 + khalf;
            v2f a, bb;
            a.x  = tile[kslot * FDIM + colbase + m];        // A: K=kslot
            a.y  = tile[(kslot + 1) * FDIM + colbase + m];  // A: K=kslot+1
            bb.x = wLds[kslot];                             // B: w[K], constant over N
            bb.y = wLds[kslot + 1];
            // 8 args: (neg_a, A, neg_b, B, c_mod, C, reuse_a, reuse_b)
            acc = __builtin_amdgcn_wmma_f32_16x16x4_f32(
                false, a, false, bb, (short)0, acc, false, false);
        }
        // D layout: lane 0 holds M=0..7 (N=0) in acc[0..7], lane 16 holds M=8..15 (N=0)
        if (lane == 0 || lane == 16) {
            const int mbase = colbase + ((lane == 16) ? 8 : 0);
            #pragma unroll
            for (int r = 0; r < 8; ++r) outLds[mbase + r] = acc[r];
        }
    }
    __syncthreads();

    // ---- coalesced scaled store ----
    out[(size_t)b * FDIM + t] = outLds[t] * invDenLds;
}

extern "C" void kernel_launch(void* const* d_in, const int* in_sizes, int n_in,
                              void* d_out, int out_size, void* d_ws, size_t ws_size,
                              hipStream_t stream) {
    const int*   nodes    = (const int*)d_in[0];   // [B]
    const int*   nbr      = (const int*)d_in[1];   // [B, 32]
    const float* features = (const float*)d_in[2]; // [N, 256]
    float*       out      = (float*)d_out;         // [B, 256]
    const int Bn = in_sizes[0];
    gnn_dedup_mean_wmma_kernel<<<Bn, 256, 0, stream>>>(nodes, nbr, features, out);
}